// Pool_TCPA_46935402610869
// MI455X (gfx1250) — compile-verified
//
#include <hip/hip_runtime.h>
#include <hip/hip_bf16.h>

typedef __attribute__((ext_vector_type(2))) float v2f;
typedef __attribute__((ext_vector_type(8))) float v8f;

#define N_TOK   197
#define DIM     768
#define N_HEADS 12
#define BATCH   128
#define P_POOL  20      // pool size (both cls and image)
#define K_TOP   5
#define L_REP   5
#define P_LEN   200     // 100 cls + 100 img prompt tokens
#define S_COLS  397     // 197 + 200

// ---------------------------------------------------------------------------
// Kernel 1: normalize the 20 key rows of the selected layer for both pools.
// Output padded to 32 rows (rows 20..31 zero) so WMMA N-tiles are clean.
// grid = 64 blocks (set = bid/32, row = bid%32), block = 256 threads.
// ---------------------------------------------------------------------------
__global__ __launch_bounds__(256)
void norm_keys_kernel(const float* __restrict__ keys_cls,
                      const float* __restrict__ keys_img,
                      const int* __restrict__ layer_ptr,
                      float* __restrict__ nk /* [2][32][768] */) {
    const int set = blockIdx.x >> 5;          // 0 = cls, 1 = img
    const int row = blockIdx.x & 31;
    const int tid = threadIdx.x;
    const int layer = *layer_ptr;

    float* dst = nk + ((size_t)set * 32 + row) * DIM;
    if (row >= P_POOL) {
        for (int j = tid; j < DIM; j += 256) dst[j] = 0.0f;
        return;
    }
    const float* src = (set ? keys_img : keys_cls) +
                       ((size_t)layer * P_POOL + row) * DIM;

    __shared__ float red[256];
    float ss = 0.0f;
    for (int j = tid; j < DIM; j += 256) { float v = src[j]; ss += v * v; }
    red[tid] = ss;
    __syncthreads();
    for (int o = 128; o > 0; o >>= 1) {
        if (tid < o) red[tid] += red[tid + o];
        __syncthreads();
    }
    const float scale = 1.0f / fmaxf(sqrtf(red[0]), 1e-12f);
    for (int j = tid; j < DIM; j += 256) dst[j] = src[j] * scale;
}

// ---------------------------------------------------------------------------
// Kernel 2: similarity GEMM (f32 WMMA 16x16x4) + top-5 routing.
// Block = 128 threads = 4 independent waves; each wave owns 16 token rows
// vs 32 key columns (20 valid) and runs 192 K=4 WMMA steps.
//
// A frag (16x4 f32): lane r (r<16) holds x[row r][k,k+1]; lane r+16 holds
// x[row r][k+2,k+3]  -> one 8B global load / lane / step.
// B frag (4x16 f32): lane n holds nk[key n][k,k+1]; lane n+16 -> [k+2,k+3].
// ---------------------------------------------------------------------------
__global__ __launch_bounds__(128)
void pool_topk_kernel(const float* __restrict__ x,
                      const float* __restrict__ nk,   // [32][768] normalized keys
                      unsigned int* __restrict__ oh,  // one 20-bit mask per row
                      float* __restrict__ dist_acc,   // scalar accumulator
                      int is_img, float inv_count) {
    __shared__ float sim[4][16][33];

    const int tid  = threadIdx.x;
    const int w    = tid >> 5;
    const int lane = tid & 31;
    const int half = lane >> 4;
    const int r    = lane & 15;

    const int gr = (blockIdx.x * 4 + w) * 16 + r;    // global row (token) id

    size_t tok_off;
    if (is_img) {
        const int b = gr / (N_TOK - 1);
        const int p = gr - b * (N_TOK - 1);
        tok_off = ((size_t)b * N_TOK + 1 + p) * DIM; // x[b, 1+p, :]
    } else {
        tok_off = (size_t)gr * N_TOK * DIM;          // x[b, 0, :]
    }

    const float* a_ptr  = x + tok_off + 2 * half;
    const float* b0_ptr = nk + (size_t)r * DIM + 2 * half;        // keys 0..15
    const float* b1_ptr = b0_ptr + (size_t)16 * DIM;              // keys 16..31

    v8f c0 = {}; v8f c1 = {};
    float ss = 0.0f;

    #pragma unroll 4
    for (int k = 0; k < DIM; k += 4) {
        v2f a  = *(const v2f*)(a_ptr  + k);
        v2f b0 = *(const v2f*)(b0_ptr + k);
        v2f b1 = *(const v2f*)(b1_ptr + k);
        ss += a.x * a.x + a.y * a.y;
        c0 = __builtin_amdgcn_wmma_f32_16x16x4_f32(false, a, false, b0,
                                                   (short)0, c0, false, false);
        c1 = __builtin_amdgcn_wmma_f32_16x16x4_f32(false, a, false, b1,
                                                   (short)0, c1, false, false);
    }

    // full row sum-of-squares: combine (k,k+1) half with (k+2,k+3) half
    ss += __shfl_xor(ss, 16, 32);
    const float scale = 1.0f / fmaxf(sqrtf(ss), 1e-12f);  // keys already unit

    // C layout: vgpr g -> row (g + 8*half), col = lane&15 (tile1 col += 16)
    const int row0 = half * 8;
    #pragma unroll
    for (int g = 0; g < 8; ++g) {
        sim[w][row0 + g][r]      = c0[g];
        sim[w][row0 + g][16 + r] = c1[g];
    }
    __syncthreads();

    float part = 0.0f;
    if (lane < 16) {
        float tv[K_TOP]; int ti[K_TOP];
        #pragma unroll
        for (int q = 0; q < K_TOP; ++q) { tv[q] = -1e30f; ti[q] = 0; }
        for (int p = 0; p < P_POOL; ++p) {
            const float s = sim[w][r][p] * scale;
            #pragma unroll
            for (int q = 0; q < K_TOP; ++q) {
                if (s > tv[q]) {
                    for (int m = K_TOP - 1; m > q; --m) { tv[m] = tv[m-1]; ti[m] = ti[m-1]; }
                    tv[q] = s; ti[q] = p;
                    break;
                }
            }
        }
        unsigned int bits = 0;
        float vsum = 0.0f;
        #pragma unroll
        for (int q = 0; q < K_TOP; ++q) { bits |= 1u << ti[q]; vsum += tv[q]; }
        oh[gr] = bits;
        part = ((float)K_TOP - vsum) * inv_count;   // sum of (1 - val), pre-scaled
    }
    // reduce lanes 0..15 into lane 0, then one atomic per wave
    #pragma unroll
    for (int o = 8; o > 0; o >>= 1) part += __shfl_down(part, o, 32);
    if (lane == 0) atomicAdd(dist_acc, part);
}

// ---------------------------------------------------------------------------
// Kernel 3: materialize mask[B, 12, 197, 397].  One block per (b, t) pair;
// row computed once, streamed out for all 12 heads.  The mask (480 MB) is
// write-once / never re-read and is 2.5x the 192 MB L2, so use non-temporal
// stores to stream it past the caches instead of thrashing L2.
// ---------------------------------------------------------------------------
__global__ __launch_bounds__(128)
void mask_fill_kernel(const unsigned int* __restrict__ oh_cls,
                      const unsigned int* __restrict__ oh_img,
                      float* __restrict__ out) {
    const int bt = blockIdx.x;
    const int b  = bt / N_TOK;
    const int t  = bt - b * N_TOK;
    const int tid = threadIdx.x;

    const unsigned int bits = (t == 0) ? oh_cls[b] : oh_img[b * (N_TOK - 1) + t - 1];
    const bool is_cls = (t == 0);

    float v[4];
    #pragma unroll
    for (int c = 0; c < 4; ++c) {
        const int j = tid + c * 128;
        float f = 0.0f;
        if (j < S_COLS) {
            if (j == 0) {
                f = 1.0f;
            } else if (j < 1 + P_LEN / 2) {              // cls-prompt columns
                f = is_cls ? (float)((bits >> ((j - 1) / L_REP)) & 1u) : 0.0f;
            } else if (j < 1 + P_LEN) {                   // img-prompt columns
                f = is_cls ? 0.0f : (float)((bits >> ((j - 101) / L_REP)) & 1u);
            } else {                                      // regular tokens
                f = 1.0f;
            }
        }
        v[c] = f;
    }

    const size_t row_base = ((size_t)b * N_HEADS * N_TOK + t) * S_COLS;
    const size_t head_stride = (size_t)N_TOK * S_COLS;
    #pragma unroll
    for (int h = 0; h < N_HEADS; ++h) {
        float* o = out + row_base + (size_t)h * head_stride;
        #pragma unroll
        for (int c = 0; c < 4; ++c) {
            const int j = tid + c * 128;
            if (j < S_COLS) __builtin_nontemporal_store(v[c], o + j);
        }
    }
}

// ---------------------------------------------------------------------------
extern "C" void kernel_launch(void* const* d_in, const int* in_sizes, int n_in,
                              void* d_out, int out_size, void* d_ws, size_t ws_size,
                              hipStream_t stream) {
    (void)in_sizes; (void)n_in; (void)out_size; (void)ws_size;

    const float* x        = (const float*)d_in[0];   // [128,197,768]
    const float* keys_cls = (const float*)d_in[1];   // [12,20,768]
    const float* keys_img = (const float*)d_in[2];   // [12,20,768]
    const int*   layer    = (const int*)d_in[3];     // scalar

    float* out = (float*)d_out;
    const size_t mask_elems = (size_t)BATCH * N_HEADS * N_TOK * S_COLS;
    float* dist = out + mask_elems;                  // trailing scalar

    // workspace layout
    float* nk = (float*)d_ws;                        // [2][32][768] f32
    unsigned int* oh_cls = (unsigned int*)(nk + 2 * 32 * DIM);   // [128]
    unsigned int* oh_img = oh_cls + BATCH;                        // [25088]

    hipMemsetAsync(dist, 0, sizeof(float), stream);

    norm_keys_kernel<<<64, 256, 0, stream>>>(keys_cls, keys_img, layer, nk);

    // cls pool: 128 rows -> 8 waves -> 2 blocks of 4 waves
    pool_topk_kernel<<<2, 128, 0, stream>>>(x, nk, oh_cls, dist,
                                            /*is_img=*/0,
                                            1.0f / (float)(BATCH * K_TOP));
    // image pool: 25088 rows -> 1568 waves -> 392 blocks
    pool_topk_kernel<<<392, 128, 0, stream>>>(x, nk + 32 * DIM, oh_img, dist,
                                              /*is_img=*/1,
                                              1.0f / (float)(BATCH * (N_TOK - 1) * K_TOP));

    mask_fill_kernel<<<BATCH * N_TOK, 128, 0, stream>>>(oh_cls, oh_img, out);
}